// SoftSplat_24438363914713
// MI455X (gfx1250) — compile-verified
//
#include <hip/hip_runtime.h>
#include <cstdint>

// Softmax splatting (softsplat, mode='soft') for N=4, C=3, H=1080, W=1920 f32.
//
// MI455X reasoning: this op is L2-atomic-throughput bound (132.7M f32 atomics
// naively vs ~15us streaming floor at 23.3 TB/s). Levers used:
//  - Interleaved [N,H,W,4] accumulator in d_ws (132.7 MB, resident in 192 MB L2).
//  - Per-block LDS pre-aggregation: 64x32 tile + 4px halo (72x40 cells = 46 KB)
//    absorbs 16 ds_add_f32/pixel; flush = 5.625 global atomics/pixel (2.8x
//    fewer L2 atomics than direct). Long-range flow (|f|>~4) -> direct atomics.
//  - Async global->LDS staging (global_load_async_to_lds_b32 + s_wait_asynccnt):
//    per-wave double-buffered pipeline for the 6 input streams, th:TH_LOAD_NT
//    so read-once data never evicts the L2-resident accumulator.
//  - WMMA inapplicable: pure scatter, no dense matmul structure.

#define EPS_F 1e-7f
#define TILE_X 64
#define TILE_Y 32
#define HALO 4
#define LTX (TILE_X + 2 * HALO)     // 72
#define LTY (TILE_Y + 2 * HALO)     // 40
#define LCELLS (LTX * LTY)          // 2880 cells * 16 B = 46.08 KB
#define NSTREAM 6
#define ROWGROUPS (TILE_Y / 4)      // 8: each thread covers 8 rows (stride 4)

// ---- CDNA5 async global->LDS copy (ASYNCcnt-tracked) ----------------------
__device__ __forceinline__ void async_g2l_b32(float* lds_dst, const float* gsrc) {
  unsigned l = (unsigned)(uintptr_t)lds_dst;               // low 32b = LDS addr
  unsigned long long g = (unsigned long long)(uintptr_t)gsrc;
  asm volatile("global_load_async_to_lds_b32 %0, %1, off th:TH_LOAD_NT"
               :: "v"(l), "v"(g) : "memory");
}
template <int N>
__device__ __forceinline__ void wait_asynccnt() {
  asm volatile("s_wait_asynccnt %0" :: "i"(N) : "memory");
}

__global__ __launch_bounds__(256)
void zero4_kernel(float4* __restrict__ p, long long n4) {
  long long i = (long long)blockIdx.x * blockDim.x + threadIdx.x;
  if (i < n4) {
    float4 z = {0.f, 0.f, 0.f, 0.f};
    p[i] = z;
  }
}

// -------- main path: async-staged, LDS-aggregated splat --------------------
__global__ __launch_bounds__(256)
void splat_lds_kernel(const float* __restrict__ tin, const float* __restrict__ flow,
                      const float* __restrict__ metric, float* __restrict__ acc,
                      int H, int W)
{
  __shared__ float4 lacc4[LCELLS];                  // 46.08 KB accumulator tile
  __shared__ float stage[8][2][NSTREAM][32];        // 12.29 KB async staging
  float* lacc = (float*)lacc4;

  const long long HW = (long long)H * W;
  const int n      = blockIdx.z;
  const int tileX0 = blockIdx.x * TILE_X;
  const int tileY0 = blockIdx.y * TILE_Y;

  {
    float4 z = {0.f, 0.f, 0.f, 0.f};
    for (int i = threadIdx.x; i < LCELLS; i += 256) lacc4[i] = z;
  }
  __syncthreads();

  const int tx   = threadIdx.x & 63;      // 0..63 (W = 30 * 64 exactly)
  const int tyB  = threadIdx.x >> 6;      // 0..3
  const int wave = threadIdx.x >> 5;      // 0..7 (wave32)
  const int lane = threadIdx.x & 31;

  const float* fBx = flow   + (size_t)n * 2 * HW;
  const float* fBy = fBx + HW;
  const float* mB  = metric + (size_t)n * HW;
  const float* iB  = tin    + (size_t)n * 3 * HW;
  float* gacc = acc + (size_t)n * HW * 4;
  const int x = tileX0 + tx;

  // issue 6 async stream loads for row-group k into buffer buf (row clamped so
  // every wave always issues exactly NSTREAM loads -> asynccnt bookkeeping holds)
  auto issue = [&](int k, int buf) {
    int y  = tileY0 + tyB + 4 * k;
    int yc = y < H ? y : H - 1;
    long long p = (long long)yc * W + x;
    async_g2l_b32(&stage[wave][buf][0][lane], fBx + p);
    async_g2l_b32(&stage[wave][buf][1][lane], fBy + p);
    async_g2l_b32(&stage[wave][buf][2][lane], mB + p);
    async_g2l_b32(&stage[wave][buf][3][lane], iB + p);
    async_g2l_b32(&stage[wave][buf][4][lane], iB + HW + p);
    async_g2l_b32(&stage[wave][buf][5][lane], iB + 2 * HW + p);
  };

  issue(0, 0);
  for (int k = 0; k < ROWGROUPS; ++k) {
    if (k + 1 < ROWGROUPS) {
      issue(k + 1, (k + 1) & 1);
      wait_asynccnt<NSTREAM>();   // async loads complete in order: group k done
    } else {
      wait_asynccnt<0>();
    }
    const int buf = k & 1;
    const int y = tileY0 + tyB + 4 * k;
    if (y < H) {
      float fx = stage[wave][buf][0][lane];
      float fy = stage[wave][buf][1][lane];
      float fltX = (float)x + fx;
      float fltY = (float)y + fy;
      // non-finite or fully out-of-range pixels contribute nothing (matches ref)
      if (__builtin_isfinite(fltX) && __builtin_isfinite(fltY) &&
          fltX > -1.0f && fltX < (float)W && fltY > -1.0f && fltY < (float)H) {
        float m  = expf(stage[wave][buf][2][lane]);
        float v0 = stage[wave][buf][3][lane] * m;
        float v1 = stage[wave][buf][4][lane] * m;
        float v2 = stage[wave][buf][5][lane] * m;

        float nwXf = floorf(fltX), nwYf = floorf(fltY);
        int ix0 = (int)nwXf, iy0 = (int)nwYf;
        float ax = fltX - nwXf, ay = fltY - nwYf;
        float wx[2] = {1.0f - ax, ax};
        float wy[2] = {1.0f - ay, ay};
#pragma unroll
        for (int ky = 0; ky < 2; ++ky) {
#pragma unroll
          for (int kx = 0; kx < 2; ++kx) {
            float w = wx[kx] * wy[ky];
            int ix = ix0 + kx, iy = iy0 + ky;
            if (w > 0.0f && ix >= 0 && ix < W && iy >= 0 && iy < H) {
              int lx = ix - tileX0 + HALO;
              int ly = iy - tileY0 + HALO;
              if ((unsigned)lx < (unsigned)LTX && (unsigned)ly < (unsigned)LTY) {
                // LDS aggregation: ds_add_f32 x4
                float* c = &lacc[(ly * LTX + lx) * 4];
                atomicAdd(c + 0, v0 * w);
                atomicAdd(c + 1, v1 * w);
                atomicAdd(c + 2, v2 * w);
                atomicAdd(c + 3, m * w);
              } else {
                // rare long-range flow: direct global atomics
                float* g = gacc + ((size_t)iy * W + ix) * 4;
                atomicAdd(g + 0, v0 * w);
                atomicAdd(g + 1, v1 * w);
                atomicAdd(g + 2, v2 * w);
                atomicAdd(g + 3, m * w);
              }
            }
          }
        }
      }
    }
  }
  __syncthreads();

  // flush LDS tile (+halo) to the global interleaved accumulator
  for (int i = threadIdx.x; i < LCELLS; i += 256) {
    float4 c = lacc4[i];
    if (c.x != 0.0f || c.y != 0.0f || c.z != 0.0f || c.w != 0.0f) {
      int ly = i / LTX;
      int lx = i - ly * LTX;
      int ix = tileX0 + lx - HALO;
      int iy = tileY0 + ly - HALO;
      if ((unsigned)ix < (unsigned)W && (unsigned)iy < (unsigned)H) {
        float* g = gacc + ((size_t)iy * W + ix) * 4;
        atomicAdd(g + 0, c.x);
        atomicAdd(g + 1, c.y);
        atomicAdd(g + 2, c.z);
        atomicAdd(g + 3, c.w);
      }
    }
  }
}

__global__ __launch_bounds__(256)
void norm_interleaved_kernel(const float4* __restrict__ acc, float* __restrict__ out,
                             int H, int W, long long total)
{
  long long t = (long long)blockIdx.x * blockDim.x + threadIdx.x;
  if (t >= total) return;
  const long long HW = (long long)H * W;
  long long n = t / HW;
  long long p = t - n * HW;
  float4 v = acc[t];                           // resident in L2
  float inv = 1.0f / (v.w + EPS_F);
  float* oB = out + (size_t)n * 3 * HW + p;
  __builtin_nontemporal_store(v.x * inv, oB);
  __builtin_nontemporal_store(v.y * inv, oB + HW);
  __builtin_nontemporal_store(v.z * inv, oB + 2 * HW);
}

// -------- fallback path (small ws): direct global atomics, planar acc ------
__global__ __launch_bounds__(256)
void splat_direct_kernel(const float* __restrict__ tin, const float* __restrict__ flow,
                         const float* __restrict__ metric, float* __restrict__ outacc,
                         float* __restrict__ wacc, int H, int W, long long total)
{
  long long t = (long long)blockIdx.x * blockDim.x + threadIdx.x;
  if (t >= total) return;
  const long long HW = (long long)H * W;
  long long n = t / HW;
  long long p = t - n * HW;
  int y = (int)(p / W);
  int x = (int)(p - (long long)y * W);

  const float* fBx = flow + (size_t)n * 2 * HW;
  float fx = __builtin_nontemporal_load(fBx + p);
  float fy = __builtin_nontemporal_load(fBx + HW + p);
  float fltX = (float)x + fx;
  float fltY = (float)y + fy;
  if (!(__builtin_isfinite(fltX) && __builtin_isfinite(fltY) &&
        fltX > -1.0f && fltX < (float)W && fltY > -1.0f && fltY < (float)H))
    return;

  const float* iB = tin + (size_t)n * 3 * HW;
  float m  = expf(__builtin_nontemporal_load(metric + (size_t)n * HW + p));
  float v0 = __builtin_nontemporal_load(iB + p) * m;
  float v1 = __builtin_nontemporal_load(iB + HW + p) * m;
  float v2 = __builtin_nontemporal_load(iB + 2 * HW + p) * m;

  float nwXf = floorf(fltX), nwYf = floorf(fltY);
  int ix0 = (int)nwXf, iy0 = (int)nwYf;
  float ax = fltX - nwXf, ay = fltY - nwYf;
  float wx[2] = {1.0f - ax, ax};
  float wy[2] = {1.0f - ay, ay};
#pragma unroll
  for (int ky = 0; ky < 2; ++ky) {
#pragma unroll
    for (int kx = 0; kx < 2; ++kx) {
      float w = wx[kx] * wy[ky];
      int ix = ix0 + kx, iy = iy0 + ky;
      if (w > 0.0f && ix >= 0 && ix < W && iy >= 0 && iy < H) {
        size_t q = (size_t)iy * W + ix;
        atomicAdd(outacc + ((size_t)n * 3 + 0) * HW + q, v0 * w);
        atomicAdd(outacc + ((size_t)n * 3 + 1) * HW + q, v1 * w);
        atomicAdd(outacc + ((size_t)n * 3 + 2) * HW + q, v2 * w);
        atomicAdd(wacc + (size_t)n * HW + q, m * w);
      }
    }
  }
}

__global__ __launch_bounds__(256)
void norm_planar_kernel(const float* __restrict__ wacc, float* __restrict__ out,
                        int H, int W, long long total)
{
  long long t = (long long)blockIdx.x * blockDim.x + threadIdx.x;
  if (t >= total) return;
  const long long HW = (long long)H * W;
  long long n = t / HW;
  long long p = t - n * HW;
  float inv = 1.0f / (wacc[t] + EPS_F);
  float* oB = out + (size_t)n * 3 * HW + p;
  oB[0]      *= inv;
  oB[HW]     *= inv;
  oB[2 * HW] *= inv;
}

extern "C" void kernel_launch(void* const* d_in, const int* in_sizes, int n_in,
                              void* d_out, int out_size, void* d_ws, size_t ws_size,
                              hipStream_t stream)
{
  const float* tin    = (const float*)d_in[0];   // [N,3,H,W]
  const float* flow   = (const float*)d_in[1];   // [N,2,H,W]
  const float* metric = (const float*)d_in[2];   // [N,1,H,W]
  float* out = (float*)d_out;                    // [N,3,H,W]

  const int N = 4, H = 1080, W = 1920;
  const long long HW  = (long long)H * W;
  const long long NHW = (long long)N * HW;       // 8,294,400
  const size_t accBytes = (size_t)NHW * 4 * sizeof(float);  // 132.7 MB

  if (ws_size >= accBytes) {
    float* acc = (float*)d_ws;
    zero4_kernel<<<(unsigned)((NHW + 255) / 256), 256, 0, stream>>>((float4*)acc, NHW);
    dim3 grid(W / TILE_X, (H + TILE_Y - 1) / TILE_Y, N);   // 30 x 34 x 4
    splat_lds_kernel<<<grid, 256, 0, stream>>>(tin, flow, metric, acc, H, W);
    norm_interleaved_kernel<<<(unsigned)((NHW + 255) / 256), 256, 0, stream>>>(
        (const float4*)acc, out, H, W, NHW);
  } else {
    // planar fallback: accumulate channels in d_out, weights in d_ws
    float* wacc = (float*)d_ws;                  // NHW floats (33.2 MB)
    long long n4o = (NHW * 3) / 4;
    long long n4w = NHW / 4;
    zero4_kernel<<<(unsigned)((n4o + 255) / 256), 256, 0, stream>>>((float4*)out, n4o);
    zero4_kernel<<<(unsigned)((n4w + 255) / 256), 256, 0, stream>>>((float4*)wacc, n4w);
    splat_direct_kernel<<<(unsigned)((NHW + 255) / 256), 256, 0, stream>>>(
        tin, flow, metric, out, wacc, H, W, NHW);
    norm_planar_kernel<<<(unsigned)((NHW + 255) / 256), 256, 0, stream>>>(
        wacc, out, H, W, NHW);
  }
}